// KVCache_27006754357438
// MI455X (gfx1250) — compile-verified
//
#include <hip/hip_runtime.h>
#include <hip/hip_bf16.h>
#include <stdint.h>

// ---------------------------------------------------------------------------
// KV-cache prefill update (quantize branch) for MI455X / gfx1250.
// Bandwidth-bound: ~210 MB moved, ~9.4 us at 23.3 TB/s. No matmul -> no WMMA;
// the CDNA5 wins here are async global<->LDS DMA for the raw copy, wave32
// ds-swizzle reductions for per-token absmax, and NT cache hints (streamed
// data, never re-read).
// ---------------------------------------------------------------------------

typedef unsigned int u32;
typedef u32 u32x4 __attribute__((ext_vector_type(4)));
typedef u32 u32x2 __attribute__((ext_vector_type(2)));
typedef int v4i __attribute__((vector_size(16)));   // builtin param type

#define AS1 __attribute__((address_space(1)))
#define AS3 __attribute__((address_space(3)))

// Problem dims (fixed by the reference).
constexpr int H        = 32;
constexpr int MAXSEQ   = 8192;
constexpr int D        = 128;
constexpr int SINK     = 4;
constexpr int WINDOW   = 1024;
constexpr int TOTAL_FP = SINK + WINDOW;      // 1028
constexpr int MID      = MAXSEQ - TOTAL_FP;  // 7164 == QUANT_CAP -> pad == 0

// Output byte offsets: d_out = concat(k_cache, v_cache, k_quant, v_quant,
// k_scale, v_scale, kv_map) flat in return order, native dtypes.
constexpr size_t N_CACHE = (size_t)H * TOTAL_FP * D;  // bf16 elems per cache
constexpr size_t N_QUANT = (size_t)H * MID * D;       // int8 elems per quant
constexpr size_t N_SCALE = (size_t)H * MID;           // f32 elems per scale
constexpr size_t OFF_KC  = 0;
constexpr size_t OFF_VC  = OFF_KC + N_CACHE * 2;
constexpr size_t OFF_KQ  = OFF_VC + N_CACHE * 2;
constexpr size_t OFF_VQ  = OFF_KQ + N_QUANT;
constexpr size_t OFF_KS  = OFF_VQ + N_QUANT;
constexpr size_t OFF_VS  = OFF_KS + N_SCALE * 4;
constexpr size_t OFF_MAP = OFF_VS + N_SCALE * 4;

// gfx1250 async global<->LDS DMA builtins (ROCm7.2/clang-22 era); guarded so
// the file compiles on toolchains that lack them (falls back to NT b128 copy).
#if defined(__has_builtin)
#if __has_builtin(__builtin_amdgcn_global_load_async_to_lds_b128) && \
    __has_builtin(__builtin_amdgcn_global_store_async_from_lds_b128)
#define USE_ASYNC_LDS 1
#endif
#endif
#ifndef USE_ASYNC_LDS
#define USE_ASYNC_LDS 0
#endif

// ---------------------------------------------------------------------------
// Kernel 1: sink+window copy.  k_cache[h, t] = k_new[h, t<4 ? t : t+7164]
// (same for v).  Both src spans are contiguous per head, so each thread just
// moves one 16B chunk.  On CDNA5 we stage it through LDS with the async DMA
// path: global_load_async_to_lds_b128 -> s_wait_asynccnt ->
// global_store_async_from_lds_b128 (each lane owns its private LDS slot, so
// no barrier is needed between its own load and store).
// ---------------------------------------------------------------------------
__global__ __launch_bounds__(256) void cache_copy_kernel(
    const u32x4* __restrict__ kin, const u32x4* __restrict__ vin,
    u32x4* __restrict__ kout, u32x4* __restrict__ vout) {
  constexpr int ROW_CHUNKS = D * 2 / 16;              // 16 x 16B per token row
  constexpr int CPT = H * TOTAL_FP * ROW_CHUNKS;      // chunks per tensor
  int c = blockIdx.x * 256 + threadIdx.x;             // [0, 2*CPT)
  int tensor = c >= CPT;
  int r = tensor ? c - CPT : c;
  int h   = r / (TOTAL_FP * ROW_CHUNKS);
  int w   = r % (TOTAL_FP * ROW_CHUNKS);
  int tok = w >> 4;                                   // output token [0,1028)
  int d16 = w & 15;                                   // chunk within row
  int tin = (tok < SINK) ? tok : tok + (MAXSEQ - WINDOW - SINK);  // +7164
  size_t src = ((size_t)h * MAXSEQ + tin) * ROW_CHUNKS + d16;
  size_t dst = ((size_t)h * TOTAL_FP + tok) * ROW_CHUNKS + d16;
  const u32x4* in  = tensor ? vin : kin;
  u32x4*       out = tensor ? vout : kout;

#if USE_ASYNC_LDS
  __shared__ __align__(16) u32x4 tile[256];
  __builtin_amdgcn_global_load_async_to_lds_b128(
      (AS1 v4i*)(in + src), (AS3 v4i*)&tile[threadIdx.x], 0, 0);
  __builtin_amdgcn_s_wait_asynccnt(0);
  __builtin_amdgcn_global_store_async_from_lds_b128(
      (AS1 v4i*)(out + dst), (AS3 v4i*)&tile[threadIdx.x], 0, 0);
  __builtin_amdgcn_s_wait_asynccnt(0);
#else
  __builtin_nontemporal_store(__builtin_nontemporal_load(in + src), out + dst);
#endif
}

// ---------------------------------------------------------------------------
// Kernel 2: per-token symmetric int8 absmax quantization of the mid region.
// One wave32 = 2 tokens; lane L owns 8 bf16 (one b128 load), 16 lanes cover
// one 256B token row.  4-step shfl_xor butterfly (masks 1,2,4,8 stay inside
// each 16-lane half) yields the row absmax in every lane.
//   scale = max(amax/127, 1e-8)   ->   inv = 127/max(amax, 127e-8)
//   q     = clamp(rint(x*inv), -127, 127)     (RNE matches jnp.round)
// ---------------------------------------------------------------------------
__global__ __launch_bounds__(256) void quant_kernel(
    const u32x4* __restrict__ kin, const u32x4* __restrict__ vin,
    u32x2* __restrict__ kq, u32x2* __restrict__ vq,
    float* __restrict__ ks, float* __restrict__ vs) {
  constexpr int PAIRS = H * MID / 2;                  // 114624 token-pairs
  int wave = blockIdx.x * 8 + (threadIdx.x >> 5);     // [0, 2*PAIRS)
  int lane = threadIdx.x & 31;
  int tensor = wave >= PAIRS;
  int p   = tensor ? wave - PAIRS : wave;
  int tok = p * 2 + (lane >> 4);                      // token in [0, H*MID)
  int sub = lane & 15;
  int h = tok / MID;
  int t = tok - h * MID;

  const u32x4* in = tensor ? vin : kin;
  size_t src = ((size_t)h * MAXSEQ + SINK + t) * 16 + sub;
  u32x4 v = __builtin_nontemporal_load(in + src);

  float f[8];
#pragma unroll
  for (int i = 0; i < 4; ++i) {                       // bf16 -> f32, exact
    u32 u = v[i];
    f[2 * i + 0] = __uint_as_float(u << 16);
    f[2 * i + 1] = __uint_as_float(u & 0xffff0000u);
  }
  float amax = 0.0f;
#pragma unroll
  for (int i = 0; i < 8; ++i) amax = fmaxf(amax, fabsf(f[i]));
#pragma unroll
  for (int m = 1; m <= 8; m <<= 1)                    // ds_swizzle butterfly
    amax = fmaxf(amax, __shfl_xor(amax, m, 32));

  float inv = 127.0f / fmaxf(amax, 1.27e-6f);         // 1/scale
  u32 pk0 = 0, pk1 = 0;
#pragma unroll
  for (int i = 0; i < 8; ++i) {
    int q = (int)rintf(f[i] * inv);
    q = q < -127 ? -127 : (q > 127 ? 127 : q);
    if (i < 4) pk0 |= (u32)(q & 0xff) << (i * 8);
    else       pk1 |= (u32)(q & 0xff) << ((i - 4) * 8);
  }
  u32x2 outv = {pk0, pk1};
  u32x2* qo = tensor ? vq : kq;
  __builtin_nontemporal_store(outv, qo + (size_t)tok * 16 + sub);
  if (sub == 0) {
    float* so = tensor ? vs : ks;
    so[tok] = fmaxf(amax * (1.0f / 127.0f), 1e-8f);
  }
}

// ---------------------------------------------------------------------------
// Kernel 3: kv_map (int64 [8192]).
//   t < 4        -> t
//   4 <= t <7168 -> -1 - (t-4)
//   t >= 7168    -> 4 + (t - 7168)
// ---------------------------------------------------------------------------
__global__ __launch_bounds__(256) void kvmap_kernel(long long* __restrict__ m) {
  int t = blockIdx.x * 256 + threadIdx.x;
  long long v;
  if (t < SINK)               v = (long long)t;
  else if (t < SINK + MID)    v = -1LL - (long long)(t - SINK);
  else                        v = (long long)(SINK + (t - (MAXSEQ - WINDOW)));
  m[t] = v;
}

extern "C" void kernel_launch(void* const* d_in, const int* in_sizes, int n_in,
                              void* d_out, int out_size, void* d_ws, size_t ws_size,
                              hipStream_t stream) {
  (void)in_sizes; (void)n_in; (void)out_size; (void)d_ws; (void)ws_size;
  const u32x4* kin = (const u32x4*)d_in[0];   // k_new  bf16 [1,32,8192,128]
  const u32x4* vin = (const u32x4*)d_in[1];   // v_new  bf16 [1,32,8192,128]
  char* out = (char*)d_out;
  u32x4*     kc = (u32x4*)(out + OFF_KC);
  u32x4*     vc = (u32x4*)(out + OFF_VC);
  u32x2*     kq = (u32x2*)(out + OFF_KQ);
  u32x2*     vq = (u32x2*)(out + OFF_VQ);
  float*     ks = (float*)(out + OFF_KS);
  float*     vs = (float*)(out + OFF_VS);
  long long* mp = (long long*)(out + OFF_MAP);

  // 2 * 32*1028*16 chunks / 256 = 4112 blocks (exact)
  cache_copy_kernel<<<4112, 256, 0, stream>>>(kin, vin, kc, vc);
  // 2 * 32*7164/2 pairs / 8 waves = 28656 blocks (exact)
  quant_kernel<<<28656, 256, 0, stream>>>(kin, vin, kq, vq, ks, vs);
  // 8192 / 256 = 32 blocks
  kvmap_kernel<<<32, 256, 0, stream>>>(mp);
}